// Model_29145648071114
// MI455X (gfx1250) — compile-verified
//
#include <hip/hip_runtime.h>

typedef __attribute__((ext_vector_type(2))) float v2f;
typedef __attribute__((ext_vector_type(8))) float v8f;

#define RULE_NUM 256
#define RES_DIM  32
#define BATCH    16384
#define ATT_DIM  16

#define ROWS_PER_BLOCK 32
#define THREADS        64   // 2 waves (wave32)

__global__ __launch_bounds__(THREADS)
void brb_kernel(const float* __restrict__ inputs,
                const float* __restrict__ AN,
                const float* __restrict__ Dp,
                const float* __restrict__ B,
                const float* __restrict__ R,
                const float* __restrict__ U,
                float* __restrict__ out)
{
    __shared__ float s_ratio[ROWS_PER_BLOCK][RULE_NUM]; // 32 KB: aw then ratio
    __shared__ float s_sb[RULE_NUM][RES_DIM];           // 32 KB: softmax(B)

    const int  tid  = threadIdx.x;
    const int  lane = tid & 31;
    const int  wid  = tid >> 5;
    const int  m    = lane & 15;
    const bool hi   = lane >= 16;

    const float dd = Dp[0] * Dp[0] + 0.1f;
    const float s2 = 1.0f / (dd * dd);        // (1/(D^2+0.1))^2

    // ---- softmax(B, axis=-1) into LDS (each thread: 4 rules) ----
    for (int r = tid; r < RULE_NUM; r += THREADS) {
        float tmp[RES_DIM];
        float mx = -3.0e38f;
        #pragma unroll
        for (int c = 0; c < RES_DIM; ++c) { tmp[c] = B[r * RES_DIM + c]; mx = fmaxf(mx, tmp[c]); }
        float sum = 0.0f;
        #pragma unroll
        for (int c = 0; c < RES_DIM; ++c) { tmp[c] = __expf(tmp[c] - mx); sum += tmp[c]; }
        const float inv = 1.0f / sum;
        #pragma unroll
        for (int c = 0; c < RES_DIM; ++c) s_sb[r][c] = tmp[c] * inv;
    }

    // ---- per-row moments S1 = sum(x), S2 = sum(x^2); build A fragment ----
    // A(16x4) layout: lanes 0-15 hold K0,K1 ; lanes 16-31 hold K2,K3 (same rows M=m)
    const int rowL = wid * 16 + m;
    const int rowG = blockIdx.x * ROWS_PER_BLOCK + rowL;
    float S1 = 0.0f, S2 = 0.0f;
    #pragma unroll
    for (int a = 0; a < ATT_DIM; ++a) {
        const float x = inputs[rowG * ATT_DIM + a];
        S1 += x;
        S2 = fmaf(x, x, S2);
    }
    v2f afrag;
    afrag.x = hi ? S2   : 1.0f;   // K2 : K0
    afrag.y = hi ? 0.0f : S1;     // K3 : K1

    // ---- 16 x V_WMMA_F32_16X16X4_F32: w[16 rows][256 rules] ----
    for (int t = 0; t < 16; ++t) {
        const int   rule = t * 16 + m;
        const float an   = AN[rule];
        v2f bfrag;
        bfrag.x = hi ? s2   : (float)ATT_DIM * an * an * s2;  // K2 : K0 coefficient
        bfrag.y = hi ? 0.0f : -2.0f * s2 * an;                // K3 : K1 coefficient
        v8f cz = {};
        v8f w = __builtin_amdgcn_wmma_f32_16x16x4_f32(
            /*neg_a=*/false, afrag, /*neg_b=*/false, bfrag,
            /*c_mod=*/(short)0, cz, /*reuse_a=*/false, /*reuse_b=*/false);

        const float rr  = R[rule];
        const float rk2 = rr * rr + 0.1f;
        #pragma unroll
        for (int j = 0; j < 8; ++j) {
            const int row = wid * 16 + j + (hi ? 8 : 0);   // D layout: M=j / j+8
            s_ratio[row][rule] = __expf(-w[j]) * rk2;      // aw
        }
    }
    __syncthreads();

    // ---- sw per row, then ratio = aw/(sw-aw) in place (one thread per row) ----
    if (tid < ROWS_PER_BLOCK) {
        float sw = 0.0f;
        for (int r = 0; r < RULE_NUM; ++r) sw += s_ratio[tid][r];
        for (int r = 0; r < RULE_NUM; ++r) {
            const float a = s_ratio[tid][r];
            s_ratio[tid][r] = a / (sw - a);
        }
    }
    __syncthreads();

    // ---- evidential product: bc[row,res] = prod_r (ratio*sb + 1) - 1 ----
    // thread -> (row = tid/2, 16 consecutive res)
    const int prow    = tid >> 1;
    const int resBase = (tid & 1) * 16;
    float bc[16];
    #pragma unroll
    for (int i = 0; i < 16; ++i) bc[i] = 1.0f;
    for (int r = 0; r < RULE_NUM; ++r) {
        const float ratio = s_ratio[prow][r];
        #pragma unroll
        for (int i = 0; i < 16; ++i)
            bc[i] *= fmaf(ratio, s_sb[r][resBase + i], 1.0f);
    }

    float psum = 0.0f, pdot = 0.0f;
    #pragma unroll
    for (int i = 0; i < 16; ++i) {
        const float b = bc[i] - 1.0f;
        psum += b;
        pdot = fmaf(b, U[resBase + i], pdot);
    }
    // combine the two half-rows (adjacent lanes, same wave32)
    psum += __shfl_xor(psum, 1, 32);
    pdot += __shfl_xor(pdot, 1, 32);

    if ((tid & 1) == 0)
        out[blockIdx.x * ROWS_PER_BLOCK + prow] = pdot / psum;
}

extern "C" void kernel_launch(void* const* d_in, const int* in_sizes, int n_in,
                              void* d_out, int out_size, void* d_ws, size_t ws_size,
                              hipStream_t stream) {
    const float* inputs = (const float*)d_in[0];
    const float* AN     = (const float*)d_in[1];
    const float* Dp     = (const float*)d_in[2];
    const float* B      = (const float*)d_in[3];
    const float* R      = (const float*)d_in[4];
    const float* U      = (const float*)d_in[5];
    float* out = (float*)d_out;

    dim3 grid(BATCH / ROWS_PER_BLOCK);   // 512 blocks
    brb_kernel<<<grid, THREADS, 0, stream>>>(inputs, AN, Dp, B, R, U, out);
}